// WaveNet_55551107006729
// MI455X (gfx1250) — compile-verified
//
#include <hip/hip_runtime.h>
#include <cmath>

// ---------------------------------------------------------------------------
// WaveNet forward for MI455X (gfx1250, wave32).
// Strategy: one launch per layer (layers are sequential anyway); h ping-pong
// (2x21MB) and skip accumulator (134MB) stay resident in the 192MB L2, so the
// 40 RMW passes run at L2 bandwidth. All matmuls use V_WMMA_F32_16X16X4_F32.
// ---------------------------------------------------------------------------

#define L_IN    20476
#define L_OUT   16384
#define LSTRIDE 20476   // fixed time-stride for h buffers (lengths shrink)
#define NBATCH  8

typedef float v2f __attribute__((ext_vector_type(2)));
typedef float v8f __attribute__((ext_vector_type(8)));

__device__ __forceinline__ v8f wmma4(v2f a, v2f b, v8f c) {
  // D = A(16x4 f32) * B(4x16 f32) + C(16x16 f32)
  return __builtin_amdgcn_wmma_f32_16x16x4_f32(
      /*neg_a=*/false, a, /*neg_b=*/false, b,
      /*c_mod=*/(short)0, c, /*reuse_a=*/false, /*reuse_b=*/false);
}

__device__ __forceinline__ v8f vzero8() {
  v8f z = {0.f, 0.f, 0.f, 0.f, 0.f, 0.f, 0.f, 0.f};
  return z;
}

// ===========================================================================
// Kernel 1: h = init_w(32x256) @ x(b,256,L_IN)   -> h0(b,32,L_IN)
// One wave per (batch, 16-timestep tile). M=32 => 2 accumulators, K=256.
// ===========================================================================
__global__ __launch_bounds__(128) void k_init(const float* __restrict__ x,
                                              const float* __restrict__ iw,
                                              float* __restrict__ h,
                                              int totalTiles, int tilesPerB) {
  __shared__ float sW[32 * 257];  // padded stride 257 -> conflict-free A reads
  const int tid = threadIdx.x;
  for (int i = tid; i < 32 * 256; i += 128) {
    sW[(i >> 8) * 257 + (i & 255)] = iw[i];
  }
  __syncthreads();

  const int warp = tid >> 5, lane = tid & 31, half = lane >> 4, lq = lane & 15;
  const int wid = blockIdx.x * 4 + warp;
  if (wid >= totalTiles) return;
  const int b = wid / tilesPerB;
  const int t0 = (wid % tilesPerB) * 16;

  const float* xb = x + (size_t)b * 256 * L_IN;
  int tcl = t0 + lq; if (tcl > L_IN - 1) tcl = L_IN - 1;  // clamp, keep EXEC full

  v8f acc0 = vzero8(), acc1 = vzero8();
  for (int k = 0; k < 256; k += 4) {
    v2f bf;
    bf.x = xb[(size_t)(k + 2 * half) * L_IN + tcl];
    bf.y = xb[(size_t)(k + 2 * half + 1) * L_IN + tcl];
    v2f a0, a1;
    a0.x = sW[lq * 257 + k + 2 * half];
    a0.y = sW[lq * 257 + k + 2 * half + 1];
    a1.x = sW[(16 + lq) * 257 + k + 2 * half];
    a1.y = sW[(16 + lq) * 257 + k + 2 * half + 1];
    acc0 = wmma4(a0, bf, acc0);
    acc1 = wmma4(a1, bf, acc1);
  }
  const int t = t0 + lq;
  if (t < L_IN) {
    float* hb = h + (size_t)b * 32 * LSTRIDE;
    for (int j = 0; j < 8; j++) {
      hb[(size_t)(j + 8 * half) * LSTRIDE + t]      = acc0[j];
      hb[(size_t)(16 + j + 8 * half) * LSTRIDE + t] = acc1[j];
    }
  }
}

// ===========================================================================
// Kernel 2: one dilated gated layer.
//  a=h[t], c=h[t+d];  f=tanh(Wf0 a + Wf1 c); g=sigm(Wg0 a + Wg1 c); z=f*g
//  h_out[t] = c + Wr z ;  skip[t - skipStart] += Ws z  (for t in skip range)
// Tap-pair convs fused as one K=64 GEMM with A=[Wf0|Wf1], B=[a;c] in LDS.
// ===========================================================================
__global__ __launch_bounds__(128) void k_layer(
    const float* __restrict__ hin, float* __restrict__ hout,
    const float* __restrict__ fw, const float* __restrict__ gw,
    const float* __restrict__ rw, const float* __restrict__ sw,
    float* __restrict__ skip, int d, int Lin, int Lout, int skipStart,
    int totalTiles, int tilesPerB) {
  __shared__ float sAf[32 * 65];    // [Wf0|Wf1] padded
  __shared__ float sAg[32 * 65];    // [Wg0|Wg1] padded
  __shared__ float sAr[32 * 33];    // res_w padded
  __shared__ float sAs[256 * 33];   // skip_w padded
  __shared__ float sAB[4][64 * 16]; // per-wave [a;c] tile (B operand)
  __shared__ float sZ[4][32 * 16];  // per-wave z tile (B operand)

  const int tid = threadIdx.x;
  // fw/gw layout: [o][c][tap] (32,32,2). A[o][k]: k<32 -> tap0, else tap1.
  for (int i = tid; i < 32 * 64; i += 128) {
    int o = i >> 6, k = i & 63;
    int src = o * 64 + ((k & 31) << 1) + (k >> 5);
    sAf[o * 65 + k] = fw[src];
    sAg[o * 65 + k] = gw[src];
  }
  for (int i = tid; i < 32 * 32; i += 128)
    sAr[(i >> 5) * 33 + (i & 31)] = rw[i];
  for (int i = tid; i < 256 * 32; i += 128)
    sAs[(i >> 5) * 33 + (i & 31)] = sw[i];
  __syncthreads();

  const int warp = tid >> 5, lane = tid & 31, half = lane >> 4, lq = lane & 15;
  const int wid = blockIdx.x * 4 + warp;
  if (wid >= totalTiles) return;
  const int b = wid / tilesPerB;
  const int t0 = (wid % tilesPerB) * 16;

  const float* hb = hin + (size_t)b * 32 * LSTRIDE;
  float* tAB = sAB[warp];
  // Stage a (rows 0..31) and c (rows 32..63); clamp tail loads.
  for (int i = lane; i < 64 * 16; i += 32) {
    int r = i >> 4, c = i & 15;
    int t = t0 + c + ((r < 32) ? 0 : d);
    if (t > Lin - 1) t = Lin - 1;
    tAB[r * 16 + c] = hb[(size_t)(r & 31) * LSTRIDE + t];
  }

  // f,g GEMMs: K=64, M=32 each -> 4 accumulators.
  v8f accf0 = vzero8(), accf1 = vzero8(), accg0 = vzero8(), accg1 = vzero8();
  for (int k = 0; k < 64; k += 4) {
    v2f bf;
    bf.x = tAB[(k + 2 * half) * 16 + lq];
    bf.y = tAB[(k + 2 * half + 1) * 16 + lq];
    v2f a;
    a.x = sAf[lq * 65 + k + 2 * half];
    a.y = sAf[lq * 65 + k + 2 * half + 1];
    accf0 = wmma4(a, bf, accf0);
    a.x = sAf[(16 + lq) * 65 + k + 2 * half];
    a.y = sAf[(16 + lq) * 65 + k + 2 * half + 1];
    accf1 = wmma4(a, bf, accf1);
    a.x = sAg[lq * 65 + k + 2 * half];
    a.y = sAg[lq * 65 + k + 2 * half + 1];
    accg0 = wmma4(a, bf, accg0);
    a.x = sAg[(16 + lq) * 65 + k + 2 * half];
    a.y = sAg[(16 + lq) * 65 + k + 2 * half + 1];
    accg1 = wmma4(a, bf, accg1);
  }

  // z = tanh(f)*sigmoid(g); relayout through LDS as B operand (32 x 16).
  float* tZ = sZ[warp];
  for (int j = 0; j < 8; j++) {
    float z0 = tanhf(accf0[j]) * (1.f / (1.f + __expf(-accg0[j])));
    float z1 = tanhf(accf1[j]) * (1.f / (1.f + __expf(-accg1[j])));
    tZ[(j + 8 * half) * 16 + lq] = z0;
    tZ[(16 + j + 8 * half) * 16 + lq] = z1;
  }
  v2f bz[8];
  for (int s = 0; s < 8; s++) {
    int k = s * 4;
    bz[s].x = tZ[(k + 2 * half) * 16 + lq];
    bz[s].y = tZ[(k + 2 * half + 1) * 16 + lq];
  }

  // Residual GEMM: h_out = c + Wr z
  v8f accr0 = vzero8(), accr1 = vzero8();
  for (int s = 0; s < 8; s++) {
    int k = s * 4;
    v2f a;
    a.x = sAr[lq * 33 + k + 2 * half];
    a.y = sAr[lq * 33 + k + 2 * half + 1];
    accr0 = wmma4(a, bz[s], accr0);
    a.x = sAr[(16 + lq) * 33 + k + 2 * half];
    a.y = sAr[(16 + lq) * 33 + k + 2 * half + 1];
    accr1 = wmma4(a, bz[s], accr1);
  }
  const int t = t0 + lq;
  if (t < Lout) {
    float* ho = hout + (size_t)b * 32 * LSTRIDE;
    for (int j = 0; j < 8; j++) {
      int r0 = j + 8 * half, r1 = 16 + j + 8 * half;
      ho[(size_t)r0 * LSTRIDE + t] = tAB[(32 + r0) * 16 + lq] + accr0[j];
      ho[(size_t)r1 * LSTRIDE + t] = tAB[(32 + r1) * 16 + lq] + accr1[j];
    }
  }

  // Skip GEMM: 256x32, 16 M-tiles; RMW of L2-resident accumulator (tile-owned).
  if (t0 + 15 >= skipStart) {
    float* sk = skip + (size_t)b * 256 * L_OUT;
    const bool valid = (t >= skipStart) && (t < Lout);
    const int st = t - skipStart;
    for (int mt = 0; mt < 16; mt++) {
      v8f acc = vzero8();
      for (int s = 0; s < 8; s++) {
        int k = s * 4;
        v2f a;
        a.x = sAs[(mt * 16 + lq) * 33 + k + 2 * half];
        a.y = sAs[(mt * 16 + lq) * 33 + k + 2 * half + 1];
        acc = wmma4(a, bz[s], acc);
      }
      if (valid) {
        for (int j = 0; j < 8; j++) {
          size_t o = (size_t)(mt * 16 + j + 8 * half) * L_OUT + st;
          sk[o] += acc[j];
        }
      }
    }
  }
}

// ===========================================================================
// Kernel 3: out = E2 @ relu(E1 @ relu(skip)). Per wave: full 256x16 tile,
// e-tile bounced through LDS between the two K=256 GEMMs. L_OUT % 16 == 0.
// ===========================================================================
__global__ __launch_bounds__(128) void k_end(const float* __restrict__ skip,
                                             const float* __restrict__ e1,
                                             const float* __restrict__ e2,
                                             float* __restrict__ out,
                                             int totalTiles, int tilesPerB) {
  __shared__ float sS[4][256 * 16];
  const int tid = threadIdx.x, warp = tid >> 5, lane = tid & 31;
  const int half = lane >> 4, lq = lane & 15;
  const int wid = blockIdx.x * 4 + warp;
  if (wid >= totalTiles) return;
  const int b = wid / tilesPerB;
  const int t0 = (wid % tilesPerB) * 16;

  float* tS = sS[warp];
  const float* skb = skip + (size_t)b * 256 * L_OUT;
  for (int i = lane; i < 256 * 16; i += 32) {
    float v = skb[(size_t)(i >> 4) * L_OUT + t0 + (i & 15)];
    tS[i] = v > 0.f ? v : 0.f;
  }

  v8f acc[16];
  for (int mt = 0; mt < 16; mt++) acc[mt] = vzero8();
  for (int k = 0; k < 256; k += 4) {
    v2f bf;
    bf.x = tS[(k + 2 * half) * 16 + lq];
    bf.y = tS[(k + 2 * half + 1) * 16 + lq];
    for (int mt = 0; mt < 16; mt++) {
      v2f a;
      a.x = e1[(size_t)(mt * 16 + lq) * 256 + k + 2 * half];
      a.y = e1[(size_t)(mt * 16 + lq) * 256 + k + 2 * half + 1];
      acc[mt] = wmma4(a, bf, acc[mt]);
    }
  }
  // relu(e) back into LDS as B operand for second GEMM.
  for (int mt = 0; mt < 16; mt++)
    for (int j = 0; j < 8; j++) {
      float v = acc[mt][j];
      tS[(mt * 16 + j + 8 * half) * 16 + lq] = v > 0.f ? v : 0.f;
    }
  for (int mt = 0; mt < 16; mt++) acc[mt] = vzero8();
  for (int k = 0; k < 256; k += 4) {
    v2f bf;
    bf.x = tS[(k + 2 * half) * 16 + lq];
    bf.y = tS[(k + 2 * half + 1) * 16 + lq];
    for (int mt = 0; mt < 16; mt++) {
      v2f a;
      a.x = e2[(size_t)(mt * 16 + lq) * 256 + k + 2 * half];
      a.y = e2[(size_t)(mt * 16 + lq) * 256 + k + 2 * half + 1];
      acc[mt] = wmma4(a, bf, acc[mt]);
    }
  }
  float* ob = out + (size_t)b * 256 * L_OUT;
  for (int mt = 0; mt < 16; mt++)
    for (int j = 0; j < 8; j++)
      ob[(size_t)(mt * 16 + j + 8 * half) * L_OUT + t0 + lq] = acc[mt][j];
}

// ===========================================================================
extern "C" void kernel_launch(void* const* d_in, const int* in_sizes, int n_in,
                              void* d_out, int out_size, void* d_ws,
                              size_t ws_size, hipStream_t stream) {
  const float* x  = (const float*)d_in[0];
  const float* iw = (const float*)d_in[1];
  const float* fw = (const float*)d_in[2];
  const float* gw = (const float*)d_in[3];
  const float* rw = (const float*)d_in[4];
  const float* sw = (const float*)d_in[5];
  const float* e1 = (const float*)d_in[6];
  const float* e2 = (const float*)d_in[7];
  float* out = (float*)d_out;

  const size_t HSZ = (size_t)NBATCH * 32 * LSTRIDE;       // 5.24M floats
  float* h0   = (float*)d_ws;
  float* h1   = h0 + HSZ;
  float* skip = h1 + HSZ;                                  // 33.55M floats

  hipMemsetAsync(skip, 0, (size_t)NBATCH * 256 * L_OUT * sizeof(float), stream);

  {
    int tpb = (L_IN + 15) / 16, tot = NBATCH * tpb;
    k_init<<<(tot + 3) / 4, 128, 0, stream>>>(x, iw, h0, tot, tpb);
  }

  const float* hi = h0;
  float* ho = h1;
  int Lin = L_IN, idx = 0;
  for (int blk = 0; blk < 4; blk++) {
    for (int i = 0; i < 10; i++) {
      int d = 1 << i;
      int Lout = Lin - d;
      int tpb = (Lout + 15) / 16, tot = NBATCH * tpb;
      k_layer<<<(tot + 3) / 4, 128, 0, stream>>>(
          hi, ho, fw + (size_t)idx * 2048, gw + (size_t)idx * 2048,
          rw + (size_t)idx * 1024, sw + (size_t)idx * 8192, skip, d, Lin, Lout,
          Lout - L_OUT, tot, tpb);
      float* tmp = ho;
      ho = const_cast<float*>(hi);
      hi = tmp;
      Lin = Lout;
      idx++;
    }
  }

  {
    int tpb = L_OUT / 16, tot = NBATCH * tpb;
    k_end<<<(tot + 3) / 4, 128, 0, stream>>>(skip, e1, e2, out, tot, tpb);
  }
}